// CoAttentionLayer_50517405335639
// MI455X (gfx1250) — compile-verified
//
#include <hip/hip_runtime.h>
#include <math.h>

// CoAttentionLayer forward for MI455X (gfx1250), wave32.
// B=8, L1=L2=128, D=UNITS=300, fp32 throughout (matches reference precision).
// GEMMs use V_WMMA_F32_16X16X4_F32 (exact fp32 matrix path on CDNA5).

#define B_  8
#define L_  128
#define D_  300
#define U_  300
#define NEGBIG 1e30f

typedef float v2f __attribute__((ext_vector_type(2)));
typedef float v8f __attribute__((ext_vector_type(8)));

__device__ __forceinline__ float wave_reduce_max(float v) {
#pragma unroll
  for (int off = 16; off >= 1; off >>= 1)
    v = fmaxf(v, __shfl_xor(v, off, 32));
  return v;
}
__device__ __forceinline__ float wave_reduce_sum(float v) {
#pragma unroll
  for (int off = 16; off >= 1; off >>= 1)
    v += __shfl_xor(v, off, 32);
  return v;
}

// ---------------------------------------------------------------------------
// r1[b,i] = Q1[b,i,:] . (w1+w4) ; r2[b,j] = Q2[b,j,:] . (w2-w4)
// ---------------------------------------------------------------------------
__global__ __launch_bounds__(256) void k_rvec(
    const float* __restrict__ Q1, const float* __restrict__ Q2,
    const float* __restrict__ w, float* __restrict__ r1, float* __restrict__ r2) {
  int b = blockIdx.x;
  int t = threadIdx.x;
  if (t < L_) {
    const float* q = Q1 + (b * L_ + t) * D_;
    float s = 0.f;
    for (int d = 0; d < D_; ++d) s += q[d] * (w[d] + w[3 * D_ + d]);
    r1[b * L_ + t] = s;
  } else {
    int j = t - L_;
    const float* q = Q2 + (b * L_ + j) * D_;
    float s = 0.f;
    for (int d = 0; d < D_; ++d) s += q[d] * (w[D_ + d] - w[3 * D_ + d]);
    r2[b * L_ + j] = s;
  }
}

// ---------------------------------------------------------------------------
// E[b,i,j] = tanh(r1[i] + r2[j] + sum_d Q1*w3*Q2 + sum_d |Q1-Q2|*w5 + w_bias)
// 16x16 (i,j) tile per block; Q1/Q2 tiles staged in LDS (stride 301 -> no
// bank conflicts for the column-indexed Q2 reads).
// ---------------------------------------------------------------------------
#define PQ 301
__global__ __launch_bounds__(256) void k_E(
    const float* __restrict__ Q1, const float* __restrict__ Q2,
    const float* __restrict__ w, const float* __restrict__ r1,
    const float* __restrict__ r2, const float* __restrict__ w_bias,
    float* __restrict__ E) {
  __shared__ float sq1[16 * PQ];
  __shared__ float sq2[16 * PQ];
  __shared__ float sw3[D_];
  __shared__ float sw5[D_];
  int b = blockIdx.z;
  int i0 = blockIdx.y * 16, j0 = blockIdx.x * 16;
  int t = threadIdx.x;
  for (int idx = t; idx < 16 * D_; idx += 256) {
    int r = idx / D_, c = idx - r * D_;
    sq1[r * PQ + c] = Q1[(b * L_ + i0 + r) * D_ + c];
    sq2[r * PQ + c] = Q2[(b * L_ + j0 + r) * D_ + c];
  }
  for (int idx = t; idx < D_; idx += 256) {
    sw3[idx] = w[2 * D_ + idx];
    sw5[idx] = w[4 * D_ + idx];
  }
  __syncthreads();
  int tx = t & 15, ty = t >> 4;
  const float* a = &sq1[ty * PQ];
  const float* c2 = &sq2[tx * PQ];
  float acc = 0.f;
  for (int d = 0; d < D_; ++d) {
    float x = a[d], y = c2[d];
    acc += x * sw3[d] * y + fabsf(x - y) * sw5[d];
  }
  float e = tanhf(acc + r1[b * L_ + i0 + ty] + r2[b * L_ + j0 + tx] + w_bias[0]);
  E[b * L_ * L_ + (i0 + ty) * L_ + (j0 + tx)] = e;
}

// ---------------------------------------------------------------------------
// Masked softmax over one 128-long "row" (possibly a strided column view).
// One wave (32 lanes x 4 elems) per row; 8 rows per 256-thread block.
// out is contiguous [row*128 + j] (so column-view input -> transposed output).
// ---------------------------------------------------------------------------
template <int RS, int CS>
__global__ __launch_bounds__(256) void k_softmax(
    const float* __restrict__ in, const int* __restrict__ mask,
    float* __restrict__ out) {
  int b = blockIdx.y;
  int wave = threadIdx.x >> 5, lane = threadIdx.x & 31;
  int row = blockIdx.x * 8 + wave;
  const float* src = in + b * L_ * L_;
  const int* mb = mask + b * L_;
  float v[4];
  float mx = -3.0e38f;
#pragma unroll
  for (int u = 0; u < 4; ++u) {
    int j = lane + 32 * u;
    float x = src[row * RS + j * CS] + ((float)mb[j] - 1.0f) * NEGBIG;
    v[u] = x;
    mx = fmaxf(mx, x);
  }
  mx = wave_reduce_max(mx);
  float sum = 0.f;
#pragma unroll
  for (int u = 0; u < 4; ++u) {
    v[u] = __expf(v[u] - mx);
    sum += v[u];
  }
  sum = wave_reduce_sum(sum);
  float inv = 1.0f / sum;
  float* dst = out + b * L_ * L_ + row * L_;
#pragma unroll
  for (int u = 0; u < 4; ++u) dst[lane + 32 * u] = v[u] * inv;
}

// ---------------------------------------------------------------------------
// fp32 WMMA GEMM with fully static shape: C[128,NN] = A[128,KK] x op(B)
// (+bias, tanh). One wave per 16x16 C tile, 8 waves/block cover M=128;
// blockIdx.x = N tile; blockIdx.y = batch.
// BT=0: B is KKxNN row-major. BT=1: B is NNxKK row-major (C = A.B^T).
//
// All strides are template constants so every unrolled load uses an
// immediate offset from a single per-window pointer increment (no per-
// iteration 64-bit address multiplies).
//
// Edge handling WITHOUT predication: in the 16x16x4 WMMA, B lane n only
// feeds D column n, and columns n >= NN are never stored — so we clamp the
// load address (n -> NN-1) and let those lanes compute garbage. The inner
// loop stays branch-free: 2 loads + 1 v_wmma.
//
// Fragment layout per CDNA5 ISA 7.12.2: lane&15 selects M (A) / N (B,C,D);
// lane>>4 selects K pair {0,1} vs {2,3}; C/D vgpr r -> M = r (+8 hi half).
// ---------------------------------------------------------------------------
template <int BT, int TANH, int LDA, int LDB, int LDC, int NN, int KK,
          int SA, int SB, int SC>
__global__ __launch_bounds__(256) void k_gemm_wmma(
    const float* __restrict__ A, const float* __restrict__ Bm,
    float* __restrict__ C, const float* __restrict__ bias) {
  int nt = blockIdx.x;
  int b = blockIdx.y;
  int wave = threadIdx.x >> 5;   // 0..7 -> M tile
  int lane = threadIdx.x & 31;
  int lo = lane & 15, hi = lane >> 4;
  int m = wave * 16 + lo;
  int n = nt * 16 + lo;
  int nc = (n < NN) ? n : (NN - 1);  // clamped address; unused if n >= NN

  // All base offsets are even floats (LDA/LDB/strides even, 2*hi even), so
  // the v2f loads below are naturally 8-byte aligned.
  const float* pA = A + b * SA + m * LDA + 2 * hi;
  const float* pB = BT ? (Bm + b * SB + nc * LDB + 2 * hi)
                       : (Bm + b * SB + 2 * hi * LDB + nc);

  v8f acc = {};
#pragma unroll 5
  for (int k = 0; k < KK; k += 4) {
    v2f av = *(const v2f*)(pA + k);
    v2f bv;
    if (BT) {
      bv = *(const v2f*)(pB + k);
    } else {
      bv.x = pB[k * LDB];
      bv.y = pB[(k + 1) * LDB];
    }
    acc = __builtin_amdgcn_wmma_f32_16x16x4_f32(
        false, av, false, bv, (short)0, acc, false, false);
  }

  if (n < NN) {
    float bval = TANH ? bias[n] : 0.f;
    float* Cb = C + b * SC + n;
#pragma unroll
    for (int r = 0; r < 8; ++r) {
      int mr = wave * 16 + hi * 8 + r;
      float vv = acc[r] + bval;
      if (TANH) vv = tanhf(vv);
      Cb[mr * LDC] = vv;
    }
  }
}

// ---------------------------------------------------------------------------
extern "C" void kernel_launch(void* const* d_in, const int* in_sizes, int n_in,
                              void* d_out, int out_size, void* d_ws,
                              size_t ws_size, hipStream_t stream) {
  (void)in_sizes; (void)n_in; (void)out_size; (void)ws_size;
  const float* Q1     = (const float*)d_in[0];  // (8,128,300)
  const float* Q2     = (const float*)d_in[1];  // (8,128,300)
  const int*   mask1  = (const int*)d_in[2];    // (8,128)
  const int*   mask2  = (const int*)d_in[3];    // (8,128)
  const float* kern   = (const float*)d_in[4];  // (2,300,300) -> kernel[0]
  const float* w      = (const float*)d_in[5];  // (1500,1)
  const float* bias   = (const float*)d_in[6];  // (2,300) -> bias[0]
  const float* w_bias = (const float*)d_in[7];  // (1,)

  float* out = (float*)d_out;
  float* beta   = out;                  // (8,128,300)
  float* alpha  = out + 307200;         // (8,128,300)
  float* Q1_new = out + 614400;         // (8,128,300)
  float* Q2_new = out + 921600;         // (8,128,300)

  // Workspace layout (floats)
  const int EL = L_ * L_;        // 16384
  const int QL = L_ * D_;        // 38400
  float* ws  = (float*)d_ws;
  float* r1  = ws;                       // 8*128
  float* r2  = r1  + B_ * L_;            // 8*128
  float* E   = r2  + B_ * L_;            // 8*16384
  float* bw  = E   + B_ * EL;            // 8*16384
  float* awT = bw  + B_ * EL;            // 8*16384
  float* Q1p = awT + B_ * EL;            // 8*38400
  float* Q2p = Q1p + B_ * QL;            // 8*38400
  float* F1  = Q2p + B_ * QL;            // 8*16384
  float* F2  = F1  + B_ * EL;            // 8*16384
  float* w1n = F2  + B_ * EL;            // 8*16384
  float* w2n = w1n + B_ * EL;            // 8*16384

  const int NT300 = (U_ + 15) / 16;  // 19
  const int NT128 = L_ / 16;         // 8

  // 1) rank-1 row/col vectors
  k_rvec<<<B_, 256, 0, stream>>>(Q1, Q2, w, r1, r2);

  // 2) fused E = tanh(bilinear + |diff| + rank-1 + bias)
  k_E<<<dim3(8, 8, B_), 256, 0, stream>>>(Q1, Q2, w, r1, r2, w_bias, E);

  // 3) bw = softmax_j(E, m2) ; awT = softmax_i(E, m1) stored transposed
  k_softmax<L_, 1><<<dim3(16, B_), 256, 0, stream>>>(E, mask2, bw);
  k_softmax<1, L_><<<dim3(16, B_), 256, 0, stream>>>(E, mask1, awT);

  // 4) beta = bw @ Q2 ; alpha = awT @ Q1   (M=128, K=128, N=300)
  k_gemm_wmma<0, 0, L_, D_, D_, U_, L_, 16384, 38400, 38400>
      <<<dim3(NT300, B_), 256, 0, stream>>>(bw, Q2, beta, nullptr);
  k_gemm_wmma<0, 0, L_, D_, D_, U_, L_, 16384, 38400, 38400>
      <<<dim3(NT300, B_), 256, 0, stream>>>(awT, Q1, alpha, nullptr);

  // 5) Q1p/Q2p = tanh(Q @ kernel[0] + bias[0])   (M=128, K=300, N=300)
  k_gemm_wmma<0, 1, D_, U_, U_, U_, D_, 38400, 0, 38400>
      <<<dim3(NT300, B_), 256, 0, stream>>>(Q1, kern, Q1p, bias);
  k_gemm_wmma<0, 1, D_, U_, U_, U_, D_, 38400, 0, 38400>
      <<<dim3(NT300, B_), 256, 0, stream>>>(Q2, kern, Q2p, bias);

  // 6) F1 = Q1p @ Q1p^T ; F2 = Q2p @ Q2p^T   (M=N=128, K=300, NT form)
  k_gemm_wmma<1, 0, U_, U_, L_, L_, U_, 38400, 38400, 16384>
      <<<dim3(NT128, B_), 256, 0, stream>>>(Q1p, Q1p, F1, nullptr);
  k_gemm_wmma<1, 0, U_, U_, L_, L_, U_, 38400, 38400, 16384>
      <<<dim3(NT128, B_), 256, 0, stream>>>(Q2p, Q2p, F2, nullptr);

  // 7) row softmaxes of F1/F2
  k_softmax<L_, 1><<<dim3(16, B_), 256, 0, stream>>>(F1, mask1, w1n);
  k_softmax<L_, 1><<<dim3(16, B_), 256, 0, stream>>>(F2, mask2, w2n);

  // 8) Q1_new = w1n @ Q1 ; Q2_new = w2n @ Q2   (M=128, K=128, N=300)
  k_gemm_wmma<0, 0, L_, D_, D_, U_, L_, 16384, 38400, 38400>
      <<<dim3(NT300, B_), 256, 0, stream>>>(w1n, Q1, Q1_new, nullptr);
  k_gemm_wmma<0, 0, L_, D_, D_, U_, L_, 16384, 38400, 38400>
      <<<dim3(NT300, B_), 256, 0, stream>>>(w2n, Q2, Q2_new, nullptr);
}